// CRFLayer_54485955117613
// MI455X (gfx1250) — compile-verified
//
#include <hip/hip_runtime.h>
#include <stdint.h>
#include <stddef.h>

// Problem constants (match the reference: B, T, K = 512, 512, 64)
constexpr int kB = 512;
constexpr int kT = 512;
constexpr int kK = 64;

// ---------------------------------------------------------------------------
// CDNA5 async global->LDS copy (ASYNCcnt path, CDNA5 ISA).
// GV mode: per-lane global address in a VGPR pair, per-lane LDS byte address
// in the vdst VGPR. Tracked by ASYNCcnt; waited with s_wait_asynccnt.
// ---------------------------------------------------------------------------
__device__ __forceinline__ void async_b32_to_lds(void* lds_dst, const void* gsrc) {
    unsigned loff = (unsigned)(size_t)(__attribute__((address_space(3))) char*)lds_dst;
    asm volatile("global_load_async_to_lds_b32 %0, %1, off"
                 :
                 : "v"(loff), "v"(gsrc)
                 : "memory");
}

__device__ __forceinline__ void wait_async0() {
    asm volatile("s_wait_asynccnt 0" ::: "memory");
}

// ---------------------------------------------------------------------------
// One workgroup per batch element; 64 threads (2 wave32), lane == dest tag kt.
//  - transitions column trans[:, kt] held in 64 VGPRs per lane (as 16 float4)
//  - alpha[K] in LDS, read via ds_load_b128 broadcasts
//  - pot row double-buffered in LDS via async global->LDS prefetch
//  - backpointers (511 x 64 u8 = 32 KB) kept entirely in LDS (320KB/WGP)
//  - 4-way-split argmax accumulators for VALU ILP, exact first-index tiebreak
// ---------------------------------------------------------------------------
__global__ __launch_bounds__(kK) void viterbi_crf_kernel(
    const float* __restrict__ inputs,   // [B, T, K]
    const float* __restrict__ trans,    // [K, K]  (from_tag, to_tag)
    int* __restrict__ out)              // [B, T] int32 tags
{
    const int b   = blockIdx.x;
    const int tid = threadIdx.x;        // kt = destination tag

    __shared__ __align__(16) float alpha[kK];
    __shared__ __align__(16) float potbuf[2][kK];
    __shared__ unsigned char    bp[(kT - 1) * kK];  // 32704 B
    __shared__ int              tags[kT];           // staging + reduction scratch

    const float* pb = inputs + (size_t)b * kT * kK;

    // ---- seq_len = trunc(mean_k(count_t(nonzero))) = total_nonzero / K ----
    // float4 scan: 4x fewer vmem ops + 4x less 64-bit address math
    int cnt = 0;
    {
        const float4* pb4 = (const float4*)pb;
        for (int j = tid; j < (kT * kK) / 4; j += kK) {
            float4 v = pb4[j];
            cnt += (v.x != 0.0f) ? 1 : 0;
            cnt += (v.y != 0.0f) ? 1 : 0;
            cnt += (v.z != 0.0f) ? 1 : 0;
            cnt += (v.w != 0.0f) ? 1 : 0;
        }
    }
    tags[tid] = cnt;
    __syncthreads();
    for (int s = kK >> 1; s > 0; s >>= 1) {
        if (tid < s) tags[tid] += tags[tid + s];
        __syncthreads();
    }
    const int seq_len = tags[0] / kK;   // int32 cast of mean (truncation)
    __syncthreads();                    // tags reused for backtrace staging

    // ---- transitions column for this lane: c4[j] = trans[4j..4j+3][kt] ----
    float4 c4[kK / 4];
#pragma unroll
    for (int j = 0; j < kK / 4; ++j) {
        c4[j].x = trans[(4 * j + 0) * kK + tid];
        c4[j].y = trans[(4 * j + 1) * kK + tid];
        c4[j].z = trans[(4 * j + 2) * kK + tid];
        c4[j].w = trans[(4 * j + 3) * kK + tid];
    }

    // ---- alpha0 = potentials at t = 0 ----
    float my_alpha = pb[tid];
    alpha[tid] = my_alpha;

    // prefetch pot row t = 1 into buffer parity 1
    async_b32_to_lds(&potbuf[1][tid], pb + kK + tid);
    __syncthreads();

    // ---- forward max-plus DP ----
    for (int t = 1; t < kT; ++t) {
        // Row t was prefetched last iteration; consume it and immediately
        // issue the prefetch for t+1 so it hides behind the compute below.
        wait_async0();
        const float pot = potbuf[t & 1][tid];
        if (t + 1 < kT)
            async_b32_to_lds(&potbuf[(t + 1) & 1][tid],
                             pb + (size_t)(t + 1) * kK + tid);

        // 4 independent (best, arg) chains -> 4x shorter VALU dep chain
        float b0 = -3.402823466e38f, b1 = b0, b2 = b0, b3 = b0;
        int   a0 = 0, a1 = 1, a2 = 2, a3 = 3;
        const float4* a4p = (const float4*)alpha;   // ds_load_b128 broadcasts
#pragma unroll
        for (int j = 0; j < kK / 4; ++j) {
            float4 s4 = a4p[j];
            s4.x += c4[j].x;  // packed-add friendly
            s4.y += c4[j].y;
            s4.z += c4[j].z;
            s4.w += c4[j].w;
            { bool g = s4.x > b0; b0 = g ? s4.x : b0; a0 = g ? 4 * j + 0 : a0; }
            { bool g = s4.y > b1; b1 = g ? s4.y : b1; a1 = g ? 4 * j + 1 : a1; }
            { bool g = s4.z > b2; b2 = g ? s4.z : b2; a2 = g ? 4 * j + 2 : a2; }
            { bool g = s4.w > b3; b3 = g ? s4.w : b3; a3 = g ? 4 * j + 3 : a3; }
        }
        // Merge with first-occurrence (smallest kf) tiebreak, like jnp.argmax
        float best = b0; int arg = a0;
        { bool g = (b1 > best) || ((b1 == best) && (a1 < arg)); best = g ? b1 : best; arg = g ? a1 : arg; }
        { bool g = (b2 > best) || ((b2 == best) && (a2 < arg)); best = g ? b2 : best; arg = g ? a2 : arg; }
        { bool g = (b3 > best) || ((b3 == best) && (a3 < arg)); best = g ? b3 : best; arg = g ? a3 : arg; }

        const bool m = (t < seq_len);            // freeze past sequence end
        float         na  = m ? (pot + best) : my_alpha;
        unsigned char bpv = (unsigned char)(m ? arg : tid);  // identity bp if frozen

        __syncthreads();                 // all lanes done reading alpha
        alpha[tid]             = na;
        my_alpha               = na;
        bp[(t - 1) * kK + tid] = bpv;
        __syncthreads();                 // new alpha visible to both waves
    }

    // ---- argmax of final alpha + backtrace (LDS-latency chain) ----
    if (tid == 0) {
        float bestv = alpha[0];
        int   lt    = 0;
#pragma unroll 8
        for (int k2 = 1; k2 < kK; ++k2) {
            if (alpha[k2] > bestv) { bestv = alpha[k2]; lt = k2; }
        }
        tags[kT - 1] = lt;
        int cur = lt;
        for (int i = kT - 2; i >= 0; --i) {
            cur     = bp[i * kK + cur];
            tags[i] = cur;
        }
    }
    __syncthreads();

    // ---- coalesced int32 store of the decoded tag sequence ----
#pragma unroll
    for (int j = tid; j < kT; j += kK)
        out[(size_t)b * kT + j] = tags[j];
}

// ---------------------------------------------------------------------------
// Host-side launcher
// ---------------------------------------------------------------------------
extern "C" void kernel_launch(void* const* d_in, const int* in_sizes, int n_in,
                              void* d_out, int out_size, void* d_ws, size_t ws_size,
                              hipStream_t stream) {
    (void)in_sizes; (void)n_in; (void)d_ws; (void)ws_size; (void)out_size;
    const float* inputs      = (const float*)d_in[0];   // [B, T, K] f32
    const float* transitions = (const float*)d_in[1];   // [K, K]   f32
    int*         out         = (int*)d_out;             // [B, T]   int32

    viterbi_crf_kernel<<<dim3(kB), dim3(kK), 0, stream>>>(inputs, transitions, out);
}